// ReactDiffDynamics_55628416418500
// MI455X (gfx1250) — compile-verified
//
#include <hip/hip_runtime.h>
#include <hip/hip_bf16.h>

// Reaction-diffusion (FitzHugh-Nagumo) RHS for MI455X / gfx1250.
// HBM-bandwidth-bound stencil (268 MB traffic, ~0.4 GFLOP): optimize data path.
//  - 2 rows per block: center rows staged to LDS via gfx1250 async DMA
//    (global_load_async_to_lds_b128, ASYNCcnt); each center row doubles as the
//    vertical halo of the other, so only the outer halo rows hit the register path
//  - outer halo rows (y0-1, y0+2) loaded b128 straight to VGPRs (192MB L2 absorbs
//    the cross-block re-reads; the whole 134MB input fits in L2)
//  - outputs written with non-temporal b128 stores (streaming, never re-read)

#define NB   64
#define NH   512
#define NW   512
#define ROWS 2     // rows per block
#define TPB  128   // 4 waves (wave32); each thread owns 4 columns in each row

typedef float f32x4 __attribute__((ext_vector_type(4)));
typedef __attribute__((address_space(3))) char lds_char_t;

// Raw 32-bit LDS byte offset for the inline-asm VDST operand of async loads.
static __device__ __forceinline__ unsigned lds_byte_off(const void* p) {
    return (unsigned)(unsigned long long)(const lds_char_t*)p;
}

__global__ __launch_bounds__(TPB) void react_diff_kernel(
    const float* __restrict__ x,       // (64, 2, 512, 512)
    const float* __restrict__ params,  // (64, 3): a, b, k
    float* __restrict__ out)           // (64, 2, 512, 512)
{
    const int y0  = blockIdx.x * ROWS;   // 0, 2, ..., 510
    const int b   = blockIdx.y;          // 0..63
    const int tid = threadIdx.x;         // 0..127

    __shared__ float sU[ROWS][NW];       // center rows of U (x-shift + cross-row reuse)
    __shared__ float sV[ROWS][NW];

    const int ym = (y0 + NH - 1) & (NH - 1);   // periodic wrap (jnp.roll)
    const int yp = (y0 + ROWS) & (NH - 1);

    // ---- async DMA: 4 center rows -> LDS (GVS mode: saddr + 32-bit byte voffset) ----
    const unsigned chunkB  = (unsigned)tid * 16u;               // thread's 16B slice
    const unsigned planeUB = (unsigned)b * (2u * NH * NW * 4u); // U plane byte offset
    const unsigned planeVB = planeUB + (unsigned)(NH * NW * 4u);
    const unsigned rowB    = (unsigned)(NW * 4u);

    const unsigned gU0 = planeUB + (unsigned)y0 * rowB + chunkB;
    const unsigned gV0 = planeVB + (unsigned)y0 * rowB + chunkB;
    const unsigned lU0 = lds_byte_off(&sU[0][tid * 4]);
    const unsigned lU1 = lds_byte_off(&sU[1][tid * 4]);
    const unsigned lV0 = lds_byte_off(&sV[0][tid * 4]);
    const unsigned lV1 = lds_byte_off(&sV[1][tid * 4]);

    asm volatile("global_load_async_to_lds_b128 %0, %1, %2"
                 :: "v"(lU0), "v"(gU0),        "s"(x) : "memory");
    asm volatile("global_load_async_to_lds_b128 %0, %1, %2"
                 :: "v"(lU1), "v"(gU0 + rowB), "s"(x) : "memory");
    asm volatile("global_load_async_to_lds_b128 %0, %1, %2"
                 :: "v"(lV0), "v"(gV0),        "s"(x) : "memory");
    asm volatile("global_load_async_to_lds_b128 %0, %1, %2"
                 :: "v"(lV1), "v"(gV0 + rowB), "s"(x) : "memory");

    // ---- outer halo rows (same-x only) -> registers, b128 loads, L2-serviced ----
    const size_t fElems = (size_t)NH * NW;
    const float* Up = x + (size_t)b * 2 * fElems;
    const float* Vp = Up + fElems;
    const int x0 = tid * 4;

    const f32x4 uTop = *(const f32x4*)(Up + (size_t)ym * NW + x0);  // row y0-1
    const f32x4 uBot = *(const f32x4*)(Up + (size_t)yp * NW + x0);  // row y0+2
    const f32x4 vTop = *(const f32x4*)(Vp + (size_t)ym * NW + x0);
    const f32x4 vBot = *(const f32x4*)(Vp + (size_t)yp * NW + x0);

    const float a  = params[b * 3 + 0];
    const float bb = params[b * 3 + 1];
    const float k  = params[b * 3 + 2];
    constexpr float DX = 0.06451612903f;
    const float inv_dx2 = 1.0f / (DX * DX);   // compile-time constant (240.25)

    // publish LDS: wait for this wave's async DMA, then workgroup barrier
    asm volatile("s_wait_asynccnt 0" ::: "memory");
    __syncthreads();

    // ---- 5-point periodic Laplacian + reaction terms, both rows ----
    f32x4 du0, du1, dv0, dv1;
#pragma unroll
    for (int i = 0; i < 4; ++i) {
        const int xc = x0 + i;
        const int xl = (xc + NW - 1) & (NW - 1);
        const int xr = (xc + 1) & (NW - 1);

        const float u0 = sU[0][xc], u1 = sU[1][xc];   // centers; also each other's
        const float v0 = sV[0][xc], v1 = sV[1][xc];   // vertical neighbors

        const float lapU0 = (sU[0][xl] + sU[0][xr] + uTop[i] + u1 - 4.0f * u0) * inv_dx2;
        const float lapU1 = (sU[1][xl] + sU[1][xr] + u0 + uBot[i] - 4.0f * u1) * inv_dx2;
        const float lapV0 = (sV[0][xl] + sV[0][xr] + vTop[i] + v1 - 4.0f * v0) * inv_dx2;
        const float lapV1 = (sV[1][xl] + sV[1][xr] + v0 + vBot[i] - 4.0f * v1) * inv_dx2;

        du0[i] = a  * lapU0 + u0 - u0 * u0 * u0 - k - v0;
        du1[i] = a  * lapU1 + u1 - u1 * u1 * u1 - k - v1;
        dv0[i] = bb * lapV0 + u0 - v0;
        dv1[i] = bb * lapV1 + u1 - v1;
    }

    // ---- streaming non-temporal b128 stores ----
    float* oU0 = out + (size_t)b * 2 * fElems + (size_t)y0 * NW + x0;
    __builtin_nontemporal_store(du0, (f32x4*)oU0);
    __builtin_nontemporal_store(du1, (f32x4*)(oU0 + NW));
    __builtin_nontemporal_store(dv0, (f32x4*)(oU0 + fElems));
    __builtin_nontemporal_store(dv1, (f32x4*)(oU0 + fElems + NW));
}

extern "C" void kernel_launch(void* const* d_in, const int* in_sizes, int n_in,
                              void* d_out, int out_size, void* d_ws, size_t ws_size,
                              hipStream_t stream) {
    // setup_inputs order: d_in[0] = t (1 elem, unused), d_in[1] = x, d_in[2] = params
    const float* x      = (const float*)d_in[1];
    const float* params = (const float*)d_in[2];
    float* out          = (float*)d_out;

    dim3 grid(NH / ROWS, NB);  // (256, 64): one block per (row-pair, batch)
    react_diff_kernel<<<grid, TPB, 0, stream>>>(x, params, out);
}